// SimpleCompPCFGCharNoDistinction_29764123361290
// MI455X (gfx1250) — compile-verified
//
#include <hip/hip_runtime.h>
#include <cmath>
#include <cstddef>

// ---------------------------------------------------------------------------
// Compound-PCFG inside algorithm for MI455X (gfx1250, wave32, WMMA).
//
// Heavy math uses V_WMMA_F32_16X16X4_F32 (fp32 matrix core): operands are
// exp() values feeding a 39-level log-domain DP, so bf16 inputs (~0.4%
// per-term) would drift ~0.1 in logZ. Problem is ~19 GFLOP and L2-resident
// (beta 10MB, expG 3MB << 192MB L2) -> latency/parallelism bound, not
// math-peak bound; fp32 WMMA keeps reference accuracy for free.
//
// Restructuring: exp(Y[a,c]-m) = sum_k expL[k,a]*expR[k,c]  (rank-(w-1) GEMM)
// so both stages of each width step are WMMA GEMMs:
//   pass1: P_{b,st} = expL^T * expR     (96x96xK, per (b,st) workgroup)
//   pass2: new      = log(P * expG_T)   (M=16 batches, K=8100, N=96, per st)
// pass2 splits K over 4 wave-groups (24 waves) + LDS reduction for the serial
// tail (ns -> 1 at w=40), stores Pt/expG K-pair interleaved so each WMMA
// fragment is one b64 load, and manually unrolls K in blocks of 8 WMMAs so 16
// independent loads issue per memory-latency window (partial s_wait overlap).
// ---------------------------------------------------------------------------

#define S90 90
#define D64 64
#define NB  16
#define NN  40
#define SP  96          // S padded to 96 for 16x16 tiling
#define S2  8100        // 90*90 (reference flattening, stride 90)
#define NEGF (-1.0e9f)

typedef float v2f __attribute__((ext_vector_type(2)));
typedef float v8f __attribute__((ext_vector_type(8)));

__device__ __forceinline__ size_t beta_idx(int b, int i, int j, int A) {
  return ((size_t)((b * (NN + 1) + i) * (NN + 1) + j)) * SP + A;
}

// ---------------------------------------------------------------------- init
__global__ void k_init_beta(float* beta, int n) {
  int i = blockIdx.x * blockDim.x + threadIdx.x;
  if (i < n) beta[i] = NEGF;
}

// ---------------------------------------------------------------- root MLP
__global__ void k_root_mlp(const float* emb, const float* inW, const float* inb,
                           const float* rW1, const float* rb1,
                           const float* rW2, const float* rb2,
                           const float* outW, const float* outb, float* rootS) {
  __shared__ float h[D64], t1[D64], t2[D64], ov[S90];
  __shared__ float red[128];
  __shared__ float mS;
  int j = threadIdx.x;  // 128 threads
  if (j < D64) {
    float a = inb[j];
    for (int d = 0; d < D64; ++d) a += inW[j * D64 + d] * emb[d];
    h[j] = a;
  }
  __syncthreads();
  for (int i = 0; i < 2; ++i) {
    if (j < D64) {
      float a = rb1[i * D64 + j];
      for (int d = 0; d < D64; ++d) a += rW1[(i * D64 + j) * D64 + d] * h[d];
      t1[j] = fmaxf(a, 0.f);
    }
    __syncthreads();
    if (j < D64) {
      float a = rb2[i * D64 + j];
      for (int d = 0; d < D64; ++d) a += rW2[(i * D64 + j) * D64 + d] * t1[d];
      t2[j] = fmaxf(a, 0.f) + h[j];
    }
    __syncthreads();
    if (j < D64) h[j] = t2[j];
    __syncthreads();
  }
  float o = -INFINITY;
  if (j < S90) {
    float a = outb[j];
    for (int d = 0; d < D64; ++d) a += outW[j * D64 + d] * h[d];
    o = a;
    ov[j] = a;
  }
  red[j] = o;
  __syncthreads();
  for (int s = 64; s > 0; s >>= 1) {
    if (j < s) red[j] = fmaxf(red[j], red[j + s]);
    __syncthreads();
  }
  if (j == 0) mS = red[0];
  __syncthreads();
  red[j] = (j < S90) ? expf(o - mS) : 0.f;
  __syncthreads();
  for (int s = 64; s > 0; s >>= 1) {
    if (j < s) red[j] += red[j + s];
    __syncthreads();
  }
  float lse = mS + logf(red[0]);
  if (j < S90) rootS[j] = ov[j] - lse;
}

// --------------------------------------------------------------- split MLP
__global__ void k_split_mlp(const float* nont, const float* inW, const float* inb,
                            const float* rW1, const float* rb1,
                            const float* rW2, const float* rb2,
                            const float* outW, const float* outb,
                            float* split0, float* split1) {
  __shared__ float xr[D64], h[D64], t1[D64], t2[D64];
  int r = blockIdx.x, j = threadIdx.x;  // 64 threads
  xr[j] = nont[r * D64 + j];
  __syncthreads();
  {
    float a = inb[j];
    for (int d = 0; d < D64; ++d) a += inW[j * D64 + d] * xr[d];
    h[j] = a;
  }
  __syncthreads();
  for (int i = 0; i < 2; ++i) {
    {
      float a = rb1[i * D64 + j];
      for (int d = 0; d < D64; ++d) a += rW1[(i * D64 + j) * D64 + d] * h[d];
      t1[j] = fmaxf(a, 0.f);
    }
    __syncthreads();
    {
      float a = rb2[i * D64 + j];
      for (int d = 0; d < D64; ++d) a += rW2[(i * D64 + j) * D64 + d] * t1[d];
      t2[j] = fmaxf(a, 0.f) + h[j];
    }
    __syncthreads();
    h[j] = t2[j];
    __syncthreads();
  }
  if (j == 0) {
    float o0 = outb[0], o1 = outb[1];
    for (int d = 0; d < D64; ++d) {
      o0 += outW[d] * h[d];
      o1 += outW[D64 + d] * h[d];
    }
    float m = fmaxf(o0, o1);
    float lse = m + logf(expf(o0 - m) + expf(o1 - m));
    split0[r] = o0 - lse;
    split1[r] = o1 - lse;
  }
}

// ------------------------------------------------------------- rule logits
__global__ void k_rule_logits(const float* nont, const float* ruleW,
                              const float* ruleb, float* rl) {
  int ac = blockIdx.x * blockDim.x + threadIdx.x;
  int A = blockIdx.y;
  if (ac >= S2) return;
  float a = ruleb[ac];
  const float* wrow = ruleW + (size_t)ac * D64;
  const float* e = nont + A * D64;
  for (int d = 0; d < D64; ++d) a += e[d] * wrow[d];
  rl[(size_t)A * S2 + ac] = a;
}

__global__ void k_rule_lse(const float* rl, float* lseA) {
  __shared__ float red[256];
  __shared__ float mS;
  int A = blockIdx.x, t = threadIdx.x;
  const float* row = rl + (size_t)A * S2;
  float m = -INFINITY;
  for (int i = t; i < S2; i += 256) m = fmaxf(m, row[i]);
  red[t] = m;
  __syncthreads();
  for (int s = 128; s > 0; s >>= 1) {
    if (t < s) red[t] = fmaxf(red[t], red[t + s]);
    __syncthreads();
  }
  if (t == 0) mS = red[0];
  __syncthreads();
  float sum = 0.f;
  for (int i = t; i < S2; i += 256) sum += expf(row[i] - mS);
  red[t] = sum;
  __syncthreads();
  for (int s = 128; s > 0; s >>= 1) {
    if (t < s) red[t] += red[t + s];
    __syncthreads();
  }
  if (t == 0) lseA[A] = mS + logf(red[0]);
}

// expG pair-interleaved: expGp[(ac>>1)*192 + A*2 + (ac&1)]
//   = exp(log_softmax(rule)[A][ac] + split0[A]); A padded to 96.
__global__ void k_expG(const float* rl, const float* lseA, const float* split0,
                       float* expGp) {
  int idx = blockIdx.x * blockDim.x + threadIdx.x;
  if (idx >= S2 * SP) return;
  int ac = idx / SP, A = idx % SP;
  float v = 0.f;
  if (A < S90) v = __expf(rl[(size_t)A * S2 + ac] - lseA[A] + split0[A]);
  expGp[(size_t)(ac >> 1) * (SP * 2) + A * 2 + (ac & 1)] = v;
}

// beta[b][i][i+1][:] = x[b][i][:] + split1
__global__ void k_leaf(const float* x, const float* split1, float* beta) {
  int idx = blockIdx.x * blockDim.x + threadIdx.x;
  if (idx >= NB * NN * SP) return;
  int A = idx % SP;
  int t = idx / SP;
  int i = t % NN;
  int b = t / NN;
  float v = NEGF;
  if (A < S90) v = x[((size_t)b * NN + i) * S90 + A] + split1[A];
  beta[beta_idx(b, i, i + 1, A)] = v;
}

// ------------------------------------------------------------ width pass 1
// P_{b,st}[a][c] = sum_k exp(left[k,a]+mKR[k]-m) * exp(right[k,c]-mKR[k])
// WMMA on LDS operands; result scattered K-pair-interleaved:
//   Ptp[st*S2*16 + (ac>>1)*32 + b*2 + (ac&1)]
__global__ __launch_bounds__(256) void width_pass1(const float* __restrict__ beta,
                                                   float* __restrict__ Ptp,
                                                   float* __restrict__ mSum, int w) {
  int st = blockIdx.x, b = blockIdx.y;
  int tid = threadIdx.x;
  int Kn = w - 1;
  int Kpad = (Kn + 3) & ~3;
  __shared__ float eL[NN * SP], eR[NN * SP];
  __shared__ float mKL[NN], mKR[NN];
  __shared__ float mTot;

  // load raw spans + per-k maxima
  for (int kk = tid; kk < Kn; kk += blockDim.x) {
    const float* lrow = beta + beta_idx(b, st, st + kk + 1, 0);
    const float* rrow = beta + beta_idx(b, st + kk + 1, st + w, 0);
    float ml = -INFINITY, mr = -INFINITY;
    for (int a = 0; a < S90; ++a) {
      float v = lrow[a]; eL[kk * SP + a] = v; ml = fmaxf(ml, v);
      float u = rrow[a]; eR[kk * SP + a] = u; mr = fmaxf(mr, u);
    }
    mKL[kk] = ml; mKR[kk] = mr;
  }
  __syncthreads();
  if (tid == 0) {
    float m = -INFINITY;
    for (int kk = 0; kk < Kn; ++kk) m = fmaxf(m, mKL[kk] + mKR[kk]);
    mTot = m;
    mSum[st * NB + b] = m;
  }
  __syncthreads();
  float m = mTot;
  // in-place exp transform; zero-pad k in [Kn,Kpad) and a,c in [90,96)
  for (int idx = tid; idx < Kpad * SP; idx += blockDim.x) {
    int kk = idx / SP, a = idx % SP;
    bool valid = (kk < Kn) && (a < S90);
    eL[idx] = valid ? __expf(eL[idx] + mKR[kk] - m) : 0.f;
    eR[idx] = valid ? __expf(eR[idx] - mKR[kk]) : 0.f;
  }
  __syncthreads();

  // 6x6 output tiles of 16x16, round-robined over 8 waves (wave-uniform flow)
  int wv = tid >> 5, lane = tid & 31;
  int lm = lane & 15, hs = lane >> 4;
  for (int t = wv; t < 36; t += 8) {
    int atile = t / 6, ctile = t % 6;
    int a = atile * 16 + lm;
    int c = ctile * 16 + lm;
    v8f acc = {};
#pragma unroll 2
    for (int kb = 0; kb < Kpad; kb += 4) {
      int k0 = kb + (hs << 1);
      v2f af, bf;
      af.x = eL[k0 * SP + a];
      af.y = eL[(k0 + 1) * SP + a];
      bf.x = eR[k0 * SP + c];
      bf.y = eR[(k0 + 1) * SP + c];
      acc = __builtin_amdgcn_wmma_f32_16x16x4_f32(false, af, false, bf,
                                                  (short)0, acc, false, false);
    }
    int cc = ctile * 16 + lm;
    if (cc < S90) {
      for (int v = 0; v < 8; ++v) {
        int aa = atile * 16 + v + hs * 8;  // D row = v + (lane<16?0:8)
        if (aa < S90) {
          int ac = aa * S90 + cc;
          Ptp[(size_t)st * S2 * NB + (size_t)(ac >> 1) * (NB * 2) + b * 2 +
              (ac & 1)] = acc[v];
        }
      }
    }
  }
}

// ------------------------------------------------------------ width pass 2
// new[b][A] = m_b + log( sum_ac P_b[ac] * expG_T[ac][A] )
// One WG per st; 24 waves = 6 A-tiles x 4 K-slices; M=16 rows = batches.
// K-slice partials reduced through LDS (fp32 sums add exactly across slices).
// Inner K loop hand-unrolled in blocks of 8 WMMAs: 16 independent b64 loads
// issue together, then 8 matrix ops drain them under partial s_wait counts.
#define KSL 4
#define KCH 2028  // K chunk per slice (mult of 4); last slice gets 2016
__global__ __launch_bounds__(768) void width_pass2(float* __restrict__ beta,
                                                   const float* __restrict__ Ptp,
                                                   const float* __restrict__ mSum,
                                                   const float* __restrict__ expGp,
                                                   int w) {
  __shared__ float red[(KSL - 1) * 6 * 32 * 8];  // 18KB partial accumulators
  int st = blockIdx.x;
  int tid = threadIdx.x;
  int wv = tid >> 5, lane = tid & 31;
  int lm = lane & 15, hs = lane >> 4;
  int ks = wv / 6;        // K slice 0..3
  int at = wv % 6;        // A tile 0..5
  int Acol = at * 16 + lm;
  const float* Pp = Ptp + (size_t)st * S2 * NB;
  int kbeg = ks * KCH;
  int kend = (ks == KSL - 1) ? S2 : kbeg + KCH;
  v8f acc = {};

  int kb = kbeg;
  int kendBlk = kbeg + ((kend - kbeg) & ~31);  // whole blocks of 8 K-steps
  for (; kb < kendBlk; kb += 32) {
    v2f af[8], bf[8];
#pragma unroll
    for (int j = 0; j < 8; ++j) {
      int kp = ((kb + 4 * j) >> 1) + hs;  // K-pair index for this half-wave
      af[j] = *(const v2f*)(Pp + (size_t)kp * (NB * 2) + lm * 2);
      bf[j] = *(const v2f*)(expGp + (size_t)kp * (SP * 2) + Acol * 2);
    }
#pragma unroll
    for (int j = 0; j < 8; ++j)
      acc = __builtin_amdgcn_wmma_f32_16x16x4_f32(false, af[j], false, bf[j],
                                                  (short)0, acc, false, false);
  }
  for (; kb < kend; kb += 4) {  // <= 7 remainder K-steps
    int kp = (kb >> 1) + hs;
    v2f af = *(const v2f*)(Pp + (size_t)kp * (NB * 2) + lm * 2);
    v2f bf = *(const v2f*)(expGp + (size_t)kp * (SP * 2) + Acol * 2);
    acc = __builtin_amdgcn_wmma_f32_16x16x4_f32(false, af, false, bf,
                                                (short)0, acc, false, false);
  }

  if (ks > 0) {
    float* dst = &red[(((ks - 1) * 6 + at) * 32 + lane) * 8];
    for (int v = 0; v < 8; ++v) dst[v] = acc[v];
  }
  __syncthreads();
  if (ks == 0) {
    for (int s = 0; s < KSL - 1; ++s) {
      const float* src = &red[((s * 6 + at) * 32 + lane) * 8];
      for (int v = 0; v < 8; ++v) acc[v] += src[v];
    }
    if (Acol < S90) {
      for (int v = 0; v < 8; ++v) {
        int brow = v + hs * 8;  // D row = batch index
        float val = mSum[st * NB + brow] + __logf(acc[v]);
        beta[beta_idx(brow, st, st + w, Acol)] = val;
      }
    }
  }
}

// --------------------------------------------------------------- finalize
__global__ void k_finalize(const float* beta, const float* rootS, float* out) {
  int b = blockIdx.x, j = threadIdx.x;  // 128 threads
  __shared__ float red[128];
  __shared__ float mS;
  float v = -INFINITY;
  if (j < S90) v = beta[beta_idx(b, 0, NN, j)] + rootS[j];
  red[j] = v;
  __syncthreads();
  for (int s = 64; s > 0; s >>= 1) {
    if (j < s) red[j] = fmaxf(red[j], red[j + s]);
    __syncthreads();
  }
  if (j == 0) mS = red[0];
  __syncthreads();
  red[j] = (j < S90) ? expf(v - mS) : 0.f;
  __syncthreads();
  for (int s = 64; s > 0; s >>= 1) {
    if (j < s) red[j] += red[j + s];
    __syncthreads();
  }
  if (j == 0) out[b] = -(mS + logf(red[0]));
}

// ------------------------------------------------------------------ launch
extern "C" void kernel_launch(void* const* d_in, const int* in_sizes, int n_in,
                              void* d_out, int out_size, void* d_ws, size_t ws_size,
                              hipStream_t stream) {
  (void)in_sizes; (void)n_in; (void)out_size;
  const float* x        = (const float*)d_in[0];
  const float* nont     = (const float*)d_in[1];
  const float* root_emb = (const float*)d_in[2];
  const float* ruleW    = (const float*)d_in[3];
  const float* ruleb    = (const float*)d_in[4];
  const float* r_inW = (const float*)d_in[5],  *r_inb = (const float*)d_in[6];
  const float* r_W1  = (const float*)d_in[7],  *r_b1  = (const float*)d_in[8];
  const float* r_W2  = (const float*)d_in[9],  *r_b2  = (const float*)d_in[10];
  const float* r_oW  = (const float*)d_in[11], *r_ob  = (const float*)d_in[12];
  const float* s_inW = (const float*)d_in[13], *s_inb = (const float*)d_in[14];
  const float* s_W1  = (const float*)d_in[15], *s_b1  = (const float*)d_in[16];
  const float* s_W2  = (const float*)d_in[17], *s_b2  = (const float*)d_in[18];
  const float* s_oW  = (const float*)d_in[19], *s_ob  = (const float*)d_in[20];
  float* out = (float*)d_out;

  const size_t betaN = (size_t)NB * (NN + 1) * (NN + 1) * SP;  // 10.3 MB
  const size_t expGN = (size_t)S2 * SP;                        //  3.1 MB
  const size_t rlN   = (size_t)S90 * S2;                       //  2.9 MB
  const size_t PtN   = (size_t)(NN - 1) * S2 * NB;             // 20.2 MB

  float* ws = (float*)d_ws;
  float* beta   = ws; ws += betaN;
  float* expGp  = ws; ws += expGN;
  float* rl     = ws; ws += rlN;
  float* lseA   = ws; ws += 96;
  float* split0 = ws; ws += 96;
  float* split1 = ws; ws += 96;
  float* rootS  = ws; ws += 96;
  float* mSum   = ws; ws += (NN - 1) * NB;
  float* Ptp    = ws; ws += PtN;
  size_t need = (size_t)((char*)ws - (char*)d_ws);
  if (ws_size < need) return;  // ~36.6 MB required

  k_init_beta<<<(int)((betaN + 255) / 256), 256, 0, stream>>>(beta, (int)betaN);
  k_root_mlp<<<1, 128, 0, stream>>>(root_emb, r_inW, r_inb, r_W1, r_b1, r_W2,
                                    r_b2, r_oW, r_ob, rootS);
  k_split_mlp<<<S90, 64, 0, stream>>>(nont, s_inW, s_inb, s_W1, s_b1, s_W2,
                                      s_b2, s_oW, s_ob, split0, split1);
  k_rule_logits<<<dim3((S2 + 255) / 256, S90), 256, 0, stream>>>(nont, ruleW,
                                                                 ruleb, rl);
  k_rule_lse<<<S90, 256, 0, stream>>>(rl, lseA);
  k_expG<<<(S2 * SP + 255) / 256, 256, 0, stream>>>(rl, lseA, split0, expGp);
  k_leaf<<<(NB * NN * SP + 255) / 256, 256, 0, stream>>>(x, split1, beta);

  for (int w = 2; w <= NN; ++w) {
    int ns = NN - w + 1;
    width_pass1<<<dim3(ns, NB), 256, 0, stream>>>(beta, Ptp, mSum, w);
    width_pass2<<<ns, 768, 0, stream>>>(beta, Ptp, mSum, expGp, w);
  }
  k_finalize<<<NB, 128, 0, stream>>>(beta, rootS, out);
}